// Mo3ENet_52862457479884
// MI455X (gfx1250) — compile-verified
//
#include <hip/hip_runtime.h>
#include <cstdint>
#include <cstddef>

// ---------------------------------------------------------------------------
// Mo3ENet neighbor-list build for MI455X (gfx1250, wave32).
//
// Roofline: 67 MB of output stores vs 23.3 TB/s HBM -> ~3us floor; compute
// (~1.3 GFLOP of distances) is negligible. Strategy: all distance math in
// LDS via V_WMMA_F32_16X16X4_F32 (d^2 = |t|^2 + |c|^2 - 2 t.c, A pre-scaled
// by -2, C seeded with norm sums), then ballot/popc stream compaction per
// row (first-K-by-index semantics incl. stable invalid-index padding), then
// fully coalesced lane==slot output stores.
// ---------------------------------------------------------------------------

typedef float v2f __attribute__((ext_vector_type(2)));
typedef float v8f __attribute__((ext_vector_type(8)));

namespace cfg {
constexpr int    NB   = 256;    // batches
constexpr int    PER  = 1024;   // points per batch
constexpr int    NIN  = 256;    // "in" points (targets, also ii candidates)
constexpr int    KS   = 32;     // neighbors kept per target
constexpr float  R2   = 36.0f;  // R = 6.0
constexpr size_t EDGES = (size_t)NB * NIN * KS;  // 2,097,152
// dynamic LDS layout (floats)
constexpr int LD2_OFF  = 0;                  // d^2 [16][1024]      = 16384
constexpr int POS_OFF  = 16384;              // positions [1024][3] =  3072
constexpr int CN_OFF   = POS_OFF + 3072;     // cand norms [1024]   =  1024
constexpr int TN_OFF   = CN_OFF + 1024;      // target norms [16]   =    16
constexpr int SIDX_OFF = TN_OFF + 16;        // staging idx  8w*32  =   256
constexpr int SD_OFF   = SIDX_OFF + 256;     // staging dist 8w*32  =   256
constexpr int LDS_FLOATS = SD_OFF + 256;     // 21008 floats = 84032 B
}  // namespace cfg

__global__ __launch_bounds__(256) void mo3enet_nbr_kernel(
    const float* __restrict__ pos, float* __restrict__ out) {
  using namespace cfg;
  const int wg   = blockIdx.x;   // 0 .. NB*16-1
  const int b    = wg >> 4;      // batch id
  const int tile = wg & 15;      // 16-row target tile within batch
  const int tid  = threadIdx.x;
  const int lane = tid & 31;
  const int wv   = tid >> 5;     // wave id 0..7

  extern __shared__ float smem[];
  float* ld2  = smem + LD2_OFF;
  float* lpos = smem + POS_OFF;
  float* lcn  = smem + CN_OFF;
  float* ltn  = smem + TN_OFF;
  int*   sidx = (int*)(smem + SIDX_OFF) + wv * KS;
  float* sdst = smem + SD_OFF + wv * KS;

  // -------- Stage 0: stage this batch's 1024 positions + norms in LDS -----
  const float* posb = pos + (size_t)b * PER * 3;
  __builtin_prefetch(posb + tid * 12, 0, 0);  // global_prefetch_b8
  for (int i = tid; i < PER * 3; i += 256) lpos[i] = posb[i];
  __syncthreads();
  for (int i = tid; i < PER; i += 256) {
    float x = lpos[i * 3], y = lpos[i * 3 + 1], z = lpos[i * 3 + 2];
    lcn[i] = x * x + y * y + z * z;
  }
  __syncthreads();
  if (tid < 16) ltn[tid] = lcn[tile * 16 + tid];  // targets are points 0..255
  __syncthreads();

  // -------- Stage 1: d^2[16][1024] via V_WMMA_F32_16X16X4_F32 -------------
  // A layout (32-bit A 16x4): lanes 0-15 hold K=0,1; lanes 16-31 hold K=2,3.
  const int kp   = (lane >= 16) ? 1 : 0;
  const int hrow = lane & 15;
  const int trow = tile * 16 + hrow;
  v2f afrag;
  afrag.x = -2.0f * lpos[trow * 3 + 2 * kp];                 // x or z
  afrag.y = kp ? 0.0f : (-2.0f * lpos[trow * 3 + 1]);        // y or pad-0
  // C/D layout: VGPR r -> row M = r + 8*(lane>=16), col N = lane&15.
  v8f tnorm;
#pragma unroll
  for (int r = 0; r < 8; ++r) tnorm[r] = ltn[r + (kp ? 8 : 0)];

  for (int t = wv; t < 64; t += 8) {          // 64 column tiles of 16 cands
    const int col = t * 16 + hrow;            // candidate point 0..1023
    v2f bfrag;
    bfrag.x = lpos[col * 3 + 2 * kp];
    bfrag.y = kp ? 0.0f : lpos[col * 3 + 1];
    v8f acc = tnorm + lcn[col];               // |t|^2 + |c|^2
    v8f d2v = __builtin_amdgcn_wmma_f32_16x16x4_f32(
        /*neg_a=*/false, afrag, /*neg_b=*/false, bfrag,
        /*c_mod=*/(short)0, acc, /*reuse_a=*/false, /*reuse_b=*/false);
#pragma unroll
    for (int r = 0; r < 8; ++r) {
      const int M = r + (kp ? 8 : 0);
      float v = d2v[r];
      v = v < 0.0f ? 0.0f : v;                // guard tiny negative d^2
      ld2[M * 1024 + col] = v;
    }
  }
  __syncthreads();

  // -------- Stage 2: first-K-by-index selection, wave per 2 rows ----------
  const int goff = b * PER;
#pragma unroll
  for (int rr = 0; rr < 2; ++rr) {
    const int r  = wv * 2 + rr;               // local row 0..15
    const int ti = tile * 16 + r;             // batch-local target index
    const float* drow = ld2 + r * 1024;
#pragma unroll
    for (int ph = 0; ph < 2; ++ph) {
      const int cb = ph ? NIN : 0;            // candidate base point
      const int C  = ph ? (PER - NIN) : NIN;  // candidate count
      sdst[lane] = 0.0f;                      // padded dist = 0
      asm volatile("s_wait_dscnt 0" ::: "memory");
      // pass 1: valid candidates in index order
      int cntV = 0;
      for (int ch = 0; ch < C; ch += 32) {
        const int pt = cb + ch + lane;
        const float d2 = drow[pt];
        const bool valid = (d2 < R2) && !(ph == 0 && pt == ti);
        const unsigned bal = __builtin_amdgcn_ballot_w32(valid);
        const int slot = cntV + __popc(bal & ((1u << lane) - 1u));
        if (valid && slot < KS) { sidx[slot] = pt; sdst[slot] = __builtin_sqrtf(d2); }
        cntV += __popc(bal);
        if (cntV >= KS) break;
      }
      const int nv = cntV < KS ? cntV : KS;
      // pass 2: stable padding with invalid indices (mask=0, dist=0)
      if (nv < KS) {
        int cnt2 = nv;
        for (int ch = 0; ch < C; ch += 32) {
          const int pt = cb + ch + lane;
          const float d2 = drow[pt];
          const bool inval = !((d2 < R2) && !(ph == 0 && pt == ti));
          const unsigned bal = __builtin_amdgcn_ballot_w32(inval);
          const int slot = cnt2 + __popc(bal & ((1u << lane) - 1u));
          if (inval && slot < KS) sidx[slot] = pt;
          cnt2 += __popc(bal);
          if (cnt2 >= KS) break;
        }
      }
      asm volatile("s_wait_dscnt 0" ::: "memory");
      // coalesced output: lane == slot
      const size_t e  = ((size_t)(b * NIN + ti)) * KS + lane;
      float* o = out + (ph ? 4 * EDGES : 0);
      o[e]             = (float)(goff + sidx[lane]);        // src
      o[EDGES + e]     = (float)(goff + ti);                // tgt
      o[2 * EDGES + e] = (lane < nv) ? 1.0f : 0.0f;         // mask
      o[3 * EDGES + e] = sdst[lane];                        // dist
    }
  }
}

extern "C" void kernel_launch(void* const* d_in, const int* in_sizes, int n_in,
                              void* d_out, int out_size, void* d_ws, size_t ws_size,
                              hipStream_t stream) {
  (void)in_sizes; (void)n_in; (void)d_ws; (void)ws_size; (void)out_size;
  const float* pos = (const float*)d_in[0];  // [N,3] float32
  float* out = (float*)d_out;                // 8*EDGES floats
  const size_t shmem = (size_t)cfg::LDS_FLOATS * sizeof(float);  // 84 KB
  hipFuncSetAttribute((const void*)mo3enet_nbr_kernel,
                      hipFuncAttributeMaxDynamicSharedMemorySize, (int)shmem);
  dim3 grid(cfg::NB * (cfg::NIN / 16));      // 4096 workgroups
  mo3enet_nbr_kernel<<<grid, 256, shmem, stream>>>(pos, out);
}